// GNN_65146063946178
// MI455X (gfx1250) — compile-verified
//
#include <hip/hip_runtime.h>
#include <hip/hip_bf16.h>

// ---------------------------------------------------------------------------
// 2-layer GCN for gfx1250 (MI455X).
//  - Dense GEMMs use V_WMMA_F32_16X16X4_F32 (full fp32 precision), with a
//    branch-free software-pipelined inner loop (K-tail peeled, selects not
//    exec-masked loads) so loads overlap the matrix pipe.
//  - Edge scatter uses coalesced float4 gathers + global_atomic_add_f32
//    into L2-resident accumulators (25.6MB << 192MB L2).
// ---------------------------------------------------------------------------

typedef __attribute__((ext_vector_type(2))) float v2f;
typedef __attribute__((ext_vector_type(8))) float v8f;

#define WMMA_F32X4(A, B, C) \
    __builtin_amdgcn_wmma_f32_16x16x4_f32(false, (A), false, (B), (short)0, (C), false, false)

// ---------------- utility kernels ----------------

__global__ void fill_kernel(float* __restrict__ p, long long n, float v) {
    long long t = (long long)blockIdx.x * blockDim.x + threadIdx.x;
    if (t < n) p[t] = v;
}

// deg[dst] += 1 per edge (deg pre-filled with 1.0 for the self-loop)
__global__ void edge_deg_kernel(const long long* __restrict__ ei,
                                float* __restrict__ deg, int E) {
    int e = blockIdx.x * blockDim.x + threadIdx.x;
    if (e < E) atomicAdd(&deg[(int)ei[(long long)E + e]], 1.0f);
}

__global__ void rsqrt_kernel(float* __restrict__ d, int n) {
    int i = blockIdx.x * blockDim.x + threadIdx.x;
    if (i < n) d[i] = rsqrtf(d[i]);   // deg >= 1 always (self-loop)
}

// ---------------- GEMM 1: xW1 = x @ W1   (K=189, N=64) ----------------
// One wave computes a 16-row x 64-col tile via 4 fp32 WMMA accumulators.
// A frag (16x4 f32): lane l holds A[l&15][2*(l>>4)+{0,1}].
// B frag (4x16 f32): lane l, vgpr v holds B[v+2*(l>>4)][l&15].
struct FragSet1 { v2f a, b0, b1, b2, b3; };

__global__ __launch_bounds__(128) void gemm1_wmma(const float* __restrict__ x,
                                                  const float* __restrict__ W,   // [189,64]
                                                  float* __restrict__ out,       // [N,64]
                                                  int n_nodes) {
    const int K = 189, NF = 64;
    int wave = threadIdx.x >> 5;
    int lane = threadIdx.x & 31;
    long long row0 = ((long long)blockIdx.x * 4 + wave) * 16;
    if (row0 >= n_nodes) return;

    int m    = lane & 15;     // M index of this lane's A rows
    int half = lane >> 4;     // 0: K pair {0,1}, 1: K pair {2,3}
    int n    = lane & 15;     // N index within 16-wide tile

    long long rr = row0 + m;
    if (rr >= n_nodes) rr = n_nodes - 1;            // safe read (N%16==0 normally)
    const float* xrow = x + rr * K;

    v8f acc0 = {}, acc1 = {}, acc2 = {}, acc3 = {};

    // Branch-free fragment load for a fully in-range K block [k0, k0+4).
    auto load_frags = [&](int k0) {
        FragSet1 f;
        const float* xp = xrow + k0 + 2 * half;
        f.a.x = xp[0];
        f.a.y = xp[1];
        const float* wp = W + (long long)(k0 + 2 * half) * NF + n;
        f.b0.x = wp[0];   f.b0.y = wp[NF + 0];
        f.b1.x = wp[16];  f.b1.y = wp[NF + 16];
        f.b2.x = wp[32];  f.b2.y = wp[NF + 32];
        f.b3.x = wp[48];  f.b3.y = wp[NF + 48];
        return f;
    };

    // Software-pipelined main loop over k = 0..187 (47 unguarded steps).
    FragSet1 cur = load_frags(0);
    for (int k0 = 4; k0 < 188; k0 += 4) {
        FragSet1 nxt = load_frags(k0);     // issue next loads before consuming cur
        acc0 = WMMA_F32X4(cur.a, cur.b0, acc0);
        acc1 = WMMA_F32X4(cur.a, cur.b1, acc1);
        acc2 = WMMA_F32X4(cur.a, cur.b2, acc2);
        acc3 = WMMA_F32X4(cur.a, cur.b3, acc3);
        cur = nxt;
    }
    acc0 = WMMA_F32X4(cur.a, cur.b0, acc0);
    acc1 = WMMA_F32X4(cur.a, cur.b1, acc1);
    acc2 = WMMA_F32X4(cur.a, cur.b2, acc2);
    acc3 = WMMA_F32X4(cur.a, cur.b3, acc3);

    // Peeled tail: k = 188 only. Load unconditionally from valid addresses,
    // zero out lanes/elements beyond K with selects (no exec-mask branches).
    {
        float xa = xrow[K - 1];                       // valid for every lane
        const float* wt = W + (long long)(K - 1) * NF + n;
        float w0 = wt[0], w1 = wt[16], w2 = wt[32], w3 = wt[48];
        float z  = (half == 0) ? 1.0f : 0.0f;
        v2f a, b0, b1, b2, b3;
        a.x  = xa * z;  a.y  = 0.0f;
        b0.x = w0 * z;  b0.y = 0.0f;
        b1.x = w1 * z;  b1.y = 0.0f;
        b2.x = w2 * z;  b2.y = 0.0f;
        b3.x = w3 * z;  b3.y = 0.0f;
        acc0 = WMMA_F32X4(a, b0, acc0);
        acc1 = WMMA_F32X4(a, b1, acc1);
        acc2 = WMMA_F32X4(a, b2, acc2);
        acc3 = WMMA_F32X4(a, b3, acc3);
    }

    // C/D layout: VGPR r -> M = r + 8*half, N = lane&15 (+ tile offset)
    for (int r = 0; r < 8; ++r) {
        long long mm = row0 + r + 8 * half;
        if (mm < n_nodes) {
            float* p = out + mm * NF + n;
            p[0]  = acc0[r];
            p[16] = acc1[r];
            p[32] = acc2[r];
            p[48] = acc3[r];
        }
    }
}

// ---------------- GEMM 2: xW2 = relu(h1 + b1) @ W2   (K=64, N=32) ----------------
struct FragSet2 { float ax, ay, bxa, bya; v2f b0, b1; };

__global__ __launch_bounds__(128) void gemm2_wmma(const float* __restrict__ h1,  // [N,64]
                                                  const float* __restrict__ bias, // [64]
                                                  const float* __restrict__ W,   // [64,32]
                                                  float* __restrict__ out,       // [N,32]
                                                  int n_nodes) {
    const int K = 64, NF = 32;
    int wave = threadIdx.x >> 5;
    int lane = threadIdx.x & 31;
    long long row0 = ((long long)blockIdx.x * 4 + wave) * 16;
    if (row0 >= n_nodes) return;

    int m    = lane & 15;
    int half = lane >> 4;
    int n    = lane & 15;

    long long rr = row0 + m;
    if (rr >= n_nodes) rr = n_nodes - 1;
    const float* hrow = h1 + rr * K;

    v8f acc0 = {}, acc1 = {};

    auto load_frags = [&](int k0) {
        FragSet2 f;
        int ka = k0 + 2 * half;
        f.ax  = hrow[ka];
        f.ay  = hrow[ka + 1];
        f.bxa = bias[ka];
        f.bya = bias[ka + 1];
        const float* wp = W + (long long)ka * NF + n;
        f.b0.x = wp[0];   f.b0.y = wp[NF + 0];
        f.b1.x = wp[16];  f.b1.y = wp[NF + 16];
        return f;
    };
    auto consume = [&](const FragSet2& f) {
        v2f a;
        a.x = fmaxf(f.ax + f.bxa, 0.0f);   // fused bias + ReLU at consume time
        a.y = fmaxf(f.ay + f.bya, 0.0f);
        acc0 = WMMA_F32X4(a, f.b0, acc0);
        acc1 = WMMA_F32X4(a, f.b1, acc1);
    };

    FragSet2 cur = load_frags(0);
    for (int k0 = 4; k0 < K; k0 += 4) {    // 16 exact steps, no tail
        FragSet2 nxt = load_frags(k0);
        consume(cur);
        cur = nxt;
    }
    consume(cur);

    for (int r = 0; r < 8; ++r) {
        long long mm = row0 + r + 8 * half;
        if (mm < n_nodes) {
            float* p = out + mm * NF + n;
            p[0]  = acc0[r];
            p[16] = acc1[r];
        }
    }
}

// ---------------- edge scatter: agg[dst] += dinv[src]*dinv[dst] * msg[src] ----------------
// Work item = (edge-or-selfloop, float4 chunk). F/4 lanes cooperate per edge:
// fully coalesced 16B gathers, 4 atomic f32 adds each.
template <int F>
__global__ void scatter_add_kernel(const long long* __restrict__ ei,
                                   const float* __restrict__ dinv,
                                   const float* __restrict__ msg,   // [N,F]
                                   float* __restrict__ agg,         // [N,F] (zeroed)
                                   int E, int n_nodes) {
    const int QP = F / 4;
    long long t = (long long)blockIdx.x * blockDim.x + threadIdx.x;
    long long total = (long long)(E + n_nodes) * QP;
    if (t >= total) return;
    int idx = (int)(t / QP);
    int q   = (int)(t % QP);

    int s, d;
    if (idx < E) {
        s = (int)ei[idx];                    // edge_index[0][idx]
        d = (int)ei[(long long)E + idx];     // edge_index[1][idx]
    } else {
        s = d = idx - E;                     // self-loop
    }
    float w = dinv[s] * dinv[d];

    const float4 v = *reinterpret_cast<const float4*>(msg + (long long)s * F + q * 4);
    float* ap = agg + (long long)d * F + q * 4;
    atomicAdd(ap + 0, w * v.x);
    atomicAdd(ap + 1, w * v.y);
    atomicAdd(ap + 2, w * v.z);
    atomicAdd(ap + 3, w * v.w);
}

// ---------------- final: relu(h2+b2) @ Wfc + bfc -> log_softmax ----------------
__global__ void final_kernel(const float* __restrict__ h2,   // [N,32]
                             const float* __restrict__ b2,   // [32]
                             const float* __restrict__ Wfc,  // [32,2]
                             const float* __restrict__ bfc,  // [2]
                             float* __restrict__ out,        // [N,2]
                             int n_nodes) {
    int i = blockIdx.x * blockDim.x + threadIdx.x;
    if (i >= n_nodes) return;
    const float* hr = h2 + (long long)i * 32;
    float l0 = bfc[0], l1 = bfc[1];
    #pragma unroll
    for (int f = 0; f < 32; ++f) {
        float v = fmaxf(hr[f] + b2[f], 0.0f);
        l0 += v * Wfc[f * 2 + 0];
        l1 += v * Wfc[f * 2 + 1];
    }
    float mx  = fmaxf(l0, l1);
    float lse = mx + logf(expf(l0 - mx) + expf(l1 - mx));
    out[(long long)i * 2 + 0] = l0 - lse;
    out[(long long)i * 2 + 1] = l1 - lse;
}

// ---------------- host-side launcher ----------------
extern "C" void kernel_launch(void* const* d_in, const int* in_sizes, int n_in,
                              void* d_out, int out_size, void* d_ws, size_t ws_size,
                              hipStream_t stream) {
    const float*     x   = (const float*)d_in[0];      // [N,189] f32
    const long long* ei  = (const long long*)d_in[1];  // [2,E]   i64
    const float*     W1  = (const float*)d_in[2];      // [189,64]
    const float*     b1  = (const float*)d_in[3];      // [64]
    const float*     W2  = (const float*)d_in[4];      // [64,32]
    const float*     b2  = (const float*)d_in[5];      // [32]
    const float*     Wfc = (const float*)d_in[6];      // [32,2]
    const float*     bfc = (const float*)d_in[7];      // [2]
    float*           out = (float*)d_out;              // [N,2]

    const int N = in_sizes[0] / 189;
    const int E = in_sizes[1] / 2;

    // workspace layout (256B aligned chunks): dinv | xW1 | h1 | xW2 | h2
    float* ws = (float*)d_ws;
    auto aup = [](long long v) { return (v + 63) & ~63LL; };
    long long off = 0;
    float* dinv = ws + off; off += aup(N);
    float* xW1  = ws + off; off += aup((long long)N * 64);
    float* h1   = ws + off; off += aup((long long)N * 64);
    float* xW2  = ws + off; off += aup((long long)N * 32);
    float* h2   = ws + off; off += aup((long long)N * 32);
    (void)ws_size; (void)n_in; (void)out_size;

    const int TB = 256;

    // 1) deg = 1 (self-loop), += 1 per incoming edge, then rsqrt in place
    fill_kernel<<<(N + TB - 1) / TB, TB, 0, stream>>>(dinv, N, 1.0f);
    edge_deg_kernel<<<(E + TB - 1) / TB, TB, 0, stream>>>(ei, dinv, E);
    rsqrt_kernel<<<(N + TB - 1) / TB, TB, 0, stream>>>(dinv, N);

    // 2) xW1 = x @ W1 (WMMA fp32)
    int waves = (N + 15) / 16;
    int blocks_g = (waves + 3) / 4;
    gemm1_wmma<<<blocks_g, 128, 0, stream>>>(x, W1, xW1, N);

    // 3) h1 = scatter_add(norm * xW1[src], dst)
    long long h1n = (long long)N * 64;
    fill_kernel<<<(int)((h1n + TB - 1) / TB), TB, 0, stream>>>(h1, h1n, 0.0f);
    long long w1t = (long long)(E + N) * 16;
    scatter_add_kernel<64><<<(int)((w1t + TB - 1) / TB), TB, 0, stream>>>(ei, dinv, xW1, h1, E, N);

    // 4) xW2 = relu(h1 + b1) @ W2 (WMMA fp32, fused bias+ReLU)
    gemm2_wmma<<<blocks_g, 128, 0, stream>>>(h1, b1, W2, xW2, N);

    // 5) h2 = scatter_add(norm * xW2[src], dst)
    long long h2n = (long long)N * 32;
    fill_kernel<<<(int)((h2n + TB - 1) / TB), TB, 0, stream>>>(h2, h2n, 0.0f);
    long long w2t = (long long)(E + N) * 8;
    scatter_add_kernel<32><<<(int)((w2t + TB - 1) / TB), TB, 0, stream>>>(ei, dinv, xW2, h2, E, N);

    // 6) logits + log_softmax
    final_kernel<<<(N + TB - 1) / TB, TB, 0, stream>>>(h2, b2, Wfc, bfc, out, N);
}